// LeafBoundaryDetector_60876866453857
// MI455X (gfx1250) — compile-verified
//
#include <hip/hip_runtime.h>
#include <hip/hip_bf16.h>
#include <math.h>

// ---------------------------------------------------------------------------
// Problem constants (from reference setup_inputs)
// ---------------------------------------------------------------------------
#define BATCH 4
#define NPTS  8192
#define NFEAT 64
#define K1    67          // F + 3 (MLP input dim)
#define LDSTR 68          // padded row stride in LDS (even -> float2 aligned)
#define H1    64
#define H2    32

typedef __attribute__((ext_vector_type(2))) float v2f;
typedef __attribute__((ext_vector_type(8))) float v8f;

// ---------------------------------------------------------------------------
// Kernel 1: fused 3-layer MLP with fp32 WMMA (V_WMMA_F32_16X16X4_F32).
// 256 threads (8 waves) per block; each wave owns a 16-point tile
// -> 128 points per block, 256 blocks cover B*N = 32768 points exactly.
// Writes raw masked scores (sigmoid * leaf_mask) to workspace.
// ---------------------------------------------------------------------------
__global__ __launch_bounds__(256)
void lbd_mlp_wmma_kernel(const float* __restrict__ points,    // [B,N,3]
                         const float* __restrict__ features,  // [B,N,64]
                         const int*   __restrict__ leaf_mask, // [B,N] 0/1
                         const float* __restrict__ W1,        // [64,67]
                         const float* __restrict__ b1,        // [64]
                         const float* __restrict__ W2,        // [32,64]
                         const float* __restrict__ b2,        // [32]
                         const float* __restrict__ W3,        // [1,32]
                         const float* __restrict__ b3,        // [1]
                         float*       __restrict__ scores_ws) // [B*N]
{
    __shared__ float Xs[128 * LDSTR];   // input tile, later overwritten by H1
    __shared__ float W1s[H1 * LDSTR];   // W1, zero-padded to K=68
    __shared__ float W2s[H2 * LDSTR];   // W2 (only K<64 used)
    __shared__ float b1s[H1];
    __shared__ float b2s[H2];
    __shared__ float w3s[H2];
    __shared__ float b3s;

    const int tid  = threadIdx.x;
    const int gpt0 = blockIdx.x * 128;          // first flat point of this block

    // ---- stage features tile: 128 rows x 64 floats, as float2 ----
    for (int idx = tid; idx < 128 * (NFEAT / 2); idx += 256) {
        const int row = idx >> 5;               // 32 float2 per row
        const int c2  = idx & 31;
        const v2f v = *(const v2f*)(features + (size_t)(gpt0 + row) * NFEAT + c2 * 2);
        *(v2f*)(&Xs[row * LDSTR + c2 * 2]) = v;
    }
    // ---- stage xyz into cols 64..66, zero col 67 ----
    for (int i = tid; i < 128 * 3; i += 256) {
        const int row = i / 3, c = i - row * 3;
        Xs[row * LDSTR + NFEAT + c] = points[(size_t)(gpt0 + row) * 3 + c];
    }
    if (tid < 128) Xs[tid * LDSTR + 67] = 0.0f;

    // ---- stage W1 (zero pad col 67) ----
    for (int i = tid; i < H1 * LDSTR; i += 256) {
        const int r = i / LDSTR, c = i - r * LDSTR;
        W1s[i] = (c < K1) ? W1[r * K1 + c] : 0.0f;
    }
    // ---- stage W2 ----
    for (int i = tid; i < H2 * LDSTR; i += 256) {
        const int r = i / LDSTR, c = i - r * LDSTR;
        W2s[i] = (c < H1) ? W2[r * H1 + c] : 0.0f;
    }
    if (tid < H1) b1s[tid] = b1[tid];
    if (tid < H2) { b2s[tid] = b2[tid]; w3s[tid] = W3[tid]; }
    if (tid == 0) b3s = b3[0];
    __syncthreads();

    const int wave  = tid >> 5;
    const int lane  = tid & 31;
    const int laneM = lane & 15;
    const int hi    = lane >> 4;          // 0: K sub 0/1, 1: K sub 2/3
    const int wrow  = wave * 16;          // this wave's rows inside Xs

    // ================= Layer 1: [16 x 68] @ [68 x 64] =================
    v8f acc0 = {}, acc1 = {}, acc2t = {}, acc3 = {};
#pragma unroll 1
    for (int kc = 0; kc < 17; ++kc) {
        const int k0 = kc * 4 + hi * 2;
        const v2f a = *(const v2f*)(&Xs[(wrow + laneM) * LDSTR + k0]);
        v2f bf;
        bf = *(const v2f*)(&W1s[(0 * 16 + laneM) * LDSTR + k0]);
        acc0 = __builtin_amdgcn_wmma_f32_16x16x4_f32(false, a, false, bf, (short)0, acc0, false, false);
        bf = *(const v2f*)(&W1s[(1 * 16 + laneM) * LDSTR + k0]);
        acc1 = __builtin_amdgcn_wmma_f32_16x16x4_f32(false, a, false, bf, (short)0, acc1, false, false);
        bf = *(const v2f*)(&W1s[(2 * 16 + laneM) * LDSTR + k0]);
        acc2t = __builtin_amdgcn_wmma_f32_16x16x4_f32(false, a, false, bf, (short)0, acc2t, false, false);
        bf = *(const v2f*)(&W1s[(3 * 16 + laneM) * LDSTR + k0]);
        acc3 = __builtin_amdgcn_wmma_f32_16x16x4_f32(false, a, false, bf, (short)0, acc3, false, false);
    }

    // bias + relu, write H1 back into this wave's own Xs rows (wave-private:
    // same-wave LDS ops are in order, no barrier needed).
    {
        const float bb0 = b1s[ 0 + laneM];
        const float bb1 = b1s[16 + laneM];
        const float bb2 = b1s[32 + laneM];
        const float bb3 = b1s[48 + laneM];
#pragma unroll
        for (int r = 0; r < 8; ++r) {
            const int row = wrow + r + hi * 8;
            Xs[row * LDSTR +  0 + laneM] = fmaxf(acc0[r]  + bb0, 0.0f);
            Xs[row * LDSTR + 16 + laneM] = fmaxf(acc1[r]  + bb1, 0.0f);
            Xs[row * LDSTR + 32 + laneM] = fmaxf(acc2t[r] + bb2, 0.0f);
            Xs[row * LDSTR + 48 + laneM] = fmaxf(acc3[r]  + bb3, 0.0f);
        }
    }

    // ================= Layer 2: [16 x 64] @ [64 x 32] =================
    v8f h0 = {}, h1acc = {};
#pragma unroll 1
    for (int kc = 0; kc < 16; ++kc) {
        const int k0 = kc * 4 + hi * 2;
        const v2f a = *(const v2f*)(&Xs[(wrow + laneM) * LDSTR + k0]);
        v2f bf;
        bf = *(const v2f*)(&W2s[(0 * 16 + laneM) * LDSTR + k0]);
        h0    = __builtin_amdgcn_wmma_f32_16x16x4_f32(false, a, false, bf, (short)0, h0, false, false);
        bf = *(const v2f*)(&W2s[(1 * 16 + laneM) * LDSTR + k0]);
        h1acc = __builtin_amdgcn_wmma_f32_16x16x4_f32(false, a, false, bf, (short)0, h1acc, false, false);
    }

    // ============ Layer 3 + sigmoid + mask, via shfl reduction ============
    const float bb20 = b2s[ 0 + laneM];
    const float bb21 = b2s[16 + laneM];
    const float w3lo = w3s[laneM];
    const float w3hi = w3s[16 + laneM];
    const float bias3 = b3s;

    float myScore = 0.0f;
    int   myRow   = -1;
#pragma unroll
    for (int r = 0; r < 8; ++r) {
        // H2[row = r + 8*hi][N = lane half]; relu then dot with W3 along N
        float v = fmaxf(h0[r]    + bb20, 0.0f) * w3lo
                + fmaxf(h1acc[r] + bb21, 0.0f) * w3hi;
        // reduce across the 16-lane group (rows r / r+8 live per half-wave)
        v += __shfl_xor(v, 1, 32);
        v += __shfl_xor(v, 2, 32);
        v += __shfl_xor(v, 4, 32);
        v += __shfl_xor(v, 8, 32);
        v += bias3;
        const float s = 1.0f / (1.0f + __expf(-v));
        if (laneM == r) { myScore = s; myRow = r + 8 * hi; }
    }

    if (myRow >= 0) {
        const int gp = gpt0 + wrow + myRow;          // flat index over B*N
        const float m = (leaf_mask[gp] != 0) ? 1.0f : 0.0f;
        scores_ws[gp] = myScore * m;
    }
}

// ---------------------------------------------------------------------------
// Deterministic block reductions (1024 threads)
// ---------------------------------------------------------------------------
__device__ __forceinline__ float blockReduceF(float v, float* buf, int tid) {
    buf[tid] = v; __syncthreads();
    for (int s = 512; s > 0; s >>= 1) {
        if (tid < s) buf[tid] += buf[tid + s];
        __syncthreads();
    }
    const float r = buf[0]; __syncthreads();
    return r;
}
__device__ __forceinline__ int blockReduceI(int v, int* buf, int tid) {
    buf[tid] = v; __syncthreads();
    for (int s = 512; s > 0; s >>= 1) {
        if (tid < s) buf[tid] += buf[tid + s];
        __syncthreads();
    }
    const int r = buf[0]; __syncthreads();
    return r;
}

// ---------------------------------------------------------------------------
// Kernel 2: per-batch finalize (1 block of 1024 threads per batch).
// Emits boundary_prob and separation_confidence. Fully deterministic
// (fixed-shape LDS tree reductions, no float atomics).
// ---------------------------------------------------------------------------
__global__ __launch_bounds__(1024)
void lbd_finalize_kernel(const float* __restrict__ scores_ws, // [B*N]
                         const float* __restrict__ points,    // [B,N,3]
                         const int*   __restrict__ leaf_mask, // [B,N]
                         float*       __restrict__ out_bp,    // [B*N]
                         float*       __restrict__ out_conf)  // [B]
{
    __shared__ int   selIdx[NPTS];      // compacted selected indices (32 KB)
    __shared__ float fbuf[1024];
    __shared__ int   ibuf[1024];
    __shared__ int   scan[1024];

    const int b   = blockIdx.x;
    const int tid = threadIdx.x;
    const float* sc = scores_ws + (size_t)b * NPTS;
    const int*   mk = leaf_mask + (size_t)b * NPTS;
    const float* pp = points    + (size_t)b * NPTS * 3;

    // ---- n_leaf (integer, deterministic) ----
    int lc = 0;
    for (int i = tid; i < NPTS; i += 1024) lc += (mk[i] != 0) ? 1 : 0;
    const int n_leaf = blockReduceI(lc, ibuf, tid);
    const float n_leaf_f = (float)n_leaf;
    const bool zeroAll = (n_leaf_f < 10.0f);

    // ---- final scores + sum (for masked mean) ----
    float lsum = 0.0f;
    for (int i = tid; i < NPTS; i += 1024) {
        const float v = zeroAll ? 0.0f : sc[i];   // sc already includes mask
        out_bp[(size_t)b * NPTS + i] = v;
        lsum += v;                                 // v == x*mask
    }
    const float sum_x = blockReduceF(lsum, fbuf, tid);
    const float mean  = sum_x / fmaxf(n_leaf_f, 1.0f);

    // ---- clarity = unbiased masked variance of scores ----
    float lss = 0.0f;
    for (int i = tid; i < NPTS; i += 1024) {
        if (mk[i] != 0) {
            const float v = zeroAll ? 0.0f : sc[i];
            const float d = v - mean;
            lss += d * d;
        }
    }
    const float clarity = blockReduceF(lss, fbuf, tid) / fmaxf(n_leaf_f - 1.0f, 1.0f);

    // ---- stable compaction of selected (> 0.7) indices ----
    // each thread owns a contiguous chunk of 8 so order == index order
    const int base = tid * 8;
    int myCnt = 0;
    for (int j = 0; j < 8; ++j) {
        const float v = zeroAll ? 0.0f : sc[base + j];
        if (v > 0.7f) ++myCnt;
    }
    scan[tid] = myCnt; __syncthreads();
    for (int off = 1; off < 1024; off <<= 1) {     // Hillis-Steele inclusive
        const int add = (tid >= off) ? scan[tid - off] : 0;
        __syncthreads();
        scan[tid] += add;
        __syncthreads();
    }
    const int cnt = scan[1023];
    int pos = scan[tid] - myCnt;                   // exclusive offset
    for (int j = 0; j < 8; ++j) {
        const float v = zeroAll ? 0.0f : sc[base + j];
        if (v > 0.7f) selIdx[pos++] = base + j;
    }
    __syncthreads();

    // ---- variance of consecutive selected-point distances ----
    const int npairs = (cnt > 1) ? (cnt - 1) : 0;
    float ld = 0.0f;
    for (int j = tid; j < npairs; j += 1024) {
        const int ia = selIdx[j], ib2 = selIdx[j + 1];
        const float dx = pp[ib2 * 3 + 0] - pp[ia * 3 + 0];
        const float dy = pp[ib2 * 3 + 1] - pp[ia * 3 + 1];
        const float dz = pp[ib2 * 3 + 2] - pp[ia * 3 + 2];
        ld += sqrtf(dx * dx + dy * dy + dz * dz);
    }
    const float sum_d = blockReduceF(ld, fbuf, tid);
    const float mean_d = sum_d / fmaxf((float)npairs, 1.0f);

    float lsd = 0.0f;
    for (int j = tid; j < npairs; j += 1024) {
        const int ia = selIdx[j], ib2 = selIdx[j + 1];
        const float dx = pp[ib2 * 3 + 0] - pp[ia * 3 + 0];
        const float dy = pp[ib2 * 3 + 1] - pp[ia * 3 + 1];
        const float dz = pp[ib2 * 3 + 2] - pp[ia * 3 + 2];
        const float dd = sqrtf(dx * dx + dy * dy + dz * dz) - mean_d;
        lsd += dd * dd;
    }
    const float dvar = blockReduceF(lsd, fbuf, tid) / fmaxf((float)npairs - 1.0f, 1.0f);

    if (tid == 0) {
        float cont = 1.0f / (dvar + 1e-8f);
        cont = fminf(fmaxf(cont, 0.0f), 1.0f);
        if (cnt <= 5) cont = 0.0f;
        float conf = fminf(fmaxf(clarity * cont, 0.0f), 1.0f);
        if (n_leaf == 0) conf = 0.0f;
        out_conf[b] = conf;
    }
}

// ---------------------------------------------------------------------------
// Launch
// ---------------------------------------------------------------------------
extern "C" void kernel_launch(void* const* d_in, const int* in_sizes, int n_in,
                              void* d_out, int out_size, void* d_ws, size_t ws_size,
                              hipStream_t stream) {
    const float* points   = (const float*)d_in[0];
    const float* features = (const float*)d_in[1];
    const int*   leafmask = (const int*)  d_in[2];
    const float* W1 = (const float*)d_in[3];
    const float* b1 = (const float*)d_in[4];
    const float* W2 = (const float*)d_in[5];
    const float* b2 = (const float*)d_in[6];
    const float* W3 = (const float*)d_in[7];
    const float* b3 = (const float*)d_in[8];

    float* out = (float*)d_out;
    float* out_bp    = out;                                   // [B*N]
    float* out_feats = out + BATCH * NPTS;                    // [B*N*F]
    float* out_conf  = out + BATCH * NPTS + BATCH * NPTS * NFEAT; // [B]

    float* scores_ws = (float*)d_ws;                          // B*N floats

    // features passthrough (pure bandwidth; D2D copy on the stream)
    hipMemcpyAsync(out_feats, features,
                   (size_t)BATCH * NPTS * NFEAT * sizeof(float),
                   hipMemcpyDeviceToDevice, stream);

    // fused WMMA MLP: 256 blocks x 256 threads (8 waves -> 128 pts/block)
    lbd_mlp_wmma_kernel<<<(BATCH * NPTS) / 128, 256, 0, stream>>>(
        points, features, leafmask, W1, b1, W2, b2, W3, b3, scores_ws);

    // per-batch stats: one 1024-thread block per batch
    lbd_finalize_kernel<<<BATCH, 1024, 0, stream>>>(
        scores_ws, points, leafmask, out_bp, out_conf);
}